// lstm_38869454029654
// MI455X (gfx1250) — compile-verified
//
#include <hip/hip_runtime.h>

// ---------------------------------------------------------------------------
// Fused 2-layer LSTM (H=4, I=2, T=50) + output projection, for gfx1250.
// One wave32 processes a tile of 16 batch elements.
// Gate GEMMs use V_WMMA_F32_16X16X4_F32 (K=4 == hidden size, exact f32).
// ---------------------------------------------------------------------------

typedef float v2f __attribute__((ext_vector_type(2)));
typedef float v8f __attribute__((ext_vector_type(8)));

#define T_STEPS 50
#define BATCH   262144
#define WAVES_PER_BLOCK 8

__device__ __forceinline__ float fast_tanhf(float x) {
#if __has_builtin(__builtin_amdgcn_tanhf)
  return __builtin_amdgcn_tanhf(x);          // v_tanh_f32 (gfx1250)
#elif __has_builtin(__builtin_amdgcn_tanh_f32)
  return __builtin_amdgcn_tanh_f32(x);
#else
  float e = __expf(-2.0f * x);
  return (1.0f - e) * __builtin_amdgcn_rcpf(1.0f + e);
#endif
}

__device__ __forceinline__ float fast_sigmoid(float x) {
  return 0.5f * fast_tanhf(0.5f * x) + 0.5f;
}

__device__ __forceinline__ v8f wmma16x16x4(v2f a, v2f b, v8f c) {
  // (neg_a, A, neg_b, B, c_mod, C, reuse_a, reuse_b)
  return __builtin_amdgcn_wmma_f32_16x16x4_f32(false, a, false, b, (short)0, c,
                                               false, false);
}

__global__ __launch_bounds__(256) void lstm2_wmma_kernel(
    const float* __restrict__ x,
    const float* __restrict__ Wih0, const float* __restrict__ Whh0,
    const float* __restrict__ bih0, const float* __restrict__ bhh0,
    const float* __restrict__ Wih1, const float* __restrict__ Whh1,
    const float* __restrict__ bih1, const float* __restrict__ bhh1,
    const float* __restrict__ Wout, const float* __restrict__ bout,
    float* __restrict__ out)
{
  // Per-wave scratch: 16x16 f32 tile, row stride 17 + extra 8 for rows 8..15
  // => conflict-free scatter (D layout) and gather (A layout).
  __shared__ float lds[WAVES_PER_BLOCK][284];

  const int lane  = threadIdx.x & 31;
  const int wave  = threadIdx.x >> 5;
  const int tile  = blockIdx.x * WAVES_PER_BLOCK + wave;
  const int mbase = tile * 16;

  const int   n    = lane & 15;   // D/C column = gate index; also owned batch row
  const int   half = lane >> 4;   // A/B K-half selector, D row-half selector
  const int   ksel = half * 2;    // hidden indices owned by this lane: ksel, ksel+1
  const float zf   = half ? 0.0f : 1.0f;

  // ---- B operands (weights) in registers. B[k][n] = W[n][k];
  //      vgpr0 = B[2*half+0][n], vgpr1 = B[2*half+1][n].
  v2f Bwih0 = { Wih0[n * 2 + 0] * zf, Wih0[n * 2 + 1] * zf };  // K rows 2,3 zero-padded
  v2f Bwhh0 = { Whh0[n * 4 + ksel], Whh0[n * 4 + ksel + 1] };
  v2f Bwih1 = { Wih1[n * 4 + ksel], Wih1[n * 4 + ksel + 1] };
  v2f Bwhh1 = { Whh1[n * 4 + ksel], Whh1[n * 4 + ksel + 1] };

  // ---- C operands: bias broadcast along M (all 8 row-VGPRs identical).
  const float b0n = bih0[n] + bhh0[n];
  const float b1n = bih1[n] + bhh1[n];
  const v8f Cb0 = { b0n, b0n, b0n, b0n, b0n, b0n, b0n, b0n };
  const v8f Cb1 = { b1n, b1n, b1n, b1n, b1n, b1n, b1n, b1n };

  // LDS addressing: element (row m, col n) lives at m*17 + (m>=8 ? 8 : 0) + n.
  const int gatherBase  = n * 17 + (n >= 8 ? 8 : 0);  // this lane's batch row
  const int scatterBase = half * 144 + n;             // (r+8*half)*17 + 8*half + n
  float* const my = lds[wave];

  // x is [B, T, 2] -> one float2 per (batch,t).
  const float2* xp = reinterpret_cast<const float2*>(x) + (size_t)(mbase + n) * T_STEPS;
  const float2* wo = reinterpret_cast<const float2*>(Wout);  // [T*4] -> [T*2] float2

  v2f   h0 = {0.f, 0.f}, h1 = {0.f, 0.f};   // A operands: h[m][ksel], h[m][ksel+1]
  float c0a = 0.f, c0b = 0.f, c1a = 0.f, c1b = 0.f;
  float acc = 0.f;                          // partial of out-projection dot

  for (int t = 0; t < T_STEPS; ++t) {
    __builtin_prefetch(xp + t + 16, 0, 0);  // global_prefetch_b8, speculative
    const float2 xv = xp[t];
    const v2f xA = { xv.x * zf, xv.y * zf };  // A: K=0,1 in lanes 0-15; K=2,3 zero

    // ===== layer 0: gates = x@Wih0^T + h0@Whh0^T + (bih0+bhh0) =====
    v8f g = wmma16x16x4(xA, Bwih0, Cb0);
    g = wmma16x16x4(h0, Bwhh0, g);

    #pragma unroll
    for (int r = 0; r < 8; ++r) my[scatterBase + r * 17] = g[r];
    asm volatile("" ::: "memory");  // keep gather after scatter

    {
      const float ri0 = my[gatherBase + 0  + ksel], ri1 = my[gatherBase + 1  + ksel];
      const float rf0 = my[gatherBase + 4  + ksel], rf1 = my[gatherBase + 5  + ksel];
      const float rg0 = my[gatherBase + 8  + ksel], rg1 = my[gatherBase + 9  + ksel];
      const float ro0 = my[gatherBase + 12 + ksel], ro1 = my[gatherBase + 13 + ksel];
      asm volatile("" ::: "memory");  // keep layer-1 scatter after this gather
      c0a = fast_sigmoid(rf0) * c0a + fast_sigmoid(ri0) * fast_tanhf(rg0);
      c0b = fast_sigmoid(rf1) * c0b + fast_sigmoid(ri1) * fast_tanhf(rg1);
      h0.x = fast_sigmoid(ro0) * fast_tanhf(c0a);
      h0.y = fast_sigmoid(ro1) * fast_tanhf(c0b);
    }

    // ===== layer 1: gates = h0@Wih1^T + h1@Whh1^T + (bih1+bhh1) =====
    v8f g2 = wmma16x16x4(h0, Bwih1, Cb1);
    g2 = wmma16x16x4(h1, Bwhh1, g2);

    #pragma unroll
    for (int r = 0; r < 8; ++r) my[scatterBase + r * 17] = g2[r];
    asm volatile("" ::: "memory");

    {
      const float ri0 = my[gatherBase + 0  + ksel], ri1 = my[gatherBase + 1  + ksel];
      const float rf0 = my[gatherBase + 4  + ksel], rf1 = my[gatherBase + 5  + ksel];
      const float rg0 = my[gatherBase + 8  + ksel], rg1 = my[gatherBase + 9  + ksel];
      const float ro0 = my[gatherBase + 12 + ksel], ro1 = my[gatherBase + 13 + ksel];
      asm volatile("" ::: "memory");
      c1a = fast_sigmoid(rf0) * c1a + fast_sigmoid(ri0) * fast_tanhf(rg0);
      c1b = fast_sigmoid(rf1) * c1b + fast_sigmoid(ri1) * fast_tanhf(rg1);
      h1.x = fast_sigmoid(ro0) * fast_tanhf(c1a);
      h1.y = fast_sigmoid(ro1) * fast_tanhf(c1b);
    }

    // ===== fused output projection: acc += h1[m][k] * Wout[t*4 + k] =====
    const float2 w = wo[t * 2 + half];
    acc += h1.x * w.x + h1.y * w.y;
  }

  // Combine k={0,1} (lanes 0-15) with k={2,3} (lanes 16-31) partials.
  acc += __shfl_xor(acc, 16, 32);
  if (half == 0) {
    out[mbase + n] = fast_sigmoid(acc + bout[0]);
  }
}

extern "C" void kernel_launch(void* const* d_in, const int* in_sizes, int n_in,
                              void* d_out, int out_size, void* d_ws, size_t ws_size,
                              hipStream_t stream) {
  (void)in_sizes; (void)n_in; (void)d_ws; (void)ws_size; (void)out_size;
  const float* x    = (const float*)d_in[0];
  const float* Wih0 = (const float*)d_in[1];
  const float* Whh0 = (const float*)d_in[2];
  const float* bih0 = (const float*)d_in[3];
  const float* bhh0 = (const float*)d_in[4];
  const float* Wih1 = (const float*)d_in[5];
  const float* Whh1 = (const float*)d_in[6];
  const float* bih1 = (const float*)d_in[7];
  const float* bhh1 = (const float*)d_in[8];
  const float* Wout = (const float*)d_in[9];
  const float* bout = (const float*)d_in[10];
  float* out = (float*)d_out;

  const int tiles  = BATCH / 16;                 // 16384 waves
  const int blocks = tiles / WAVES_PER_BLOCK;    // 2048 blocks of 256 threads
  lstm2_wmma_kernel<<<blocks, 256, 0, stream>>>(
      x, Wih0, Whh0, bih0, bhh0, Wih1, Whh1, bih1, bhh1, Wout, bout, out);
}